// EpisodicMemory_56985626083905
// MI455X (gfx1250) — compile-verified
//
#include <hip/hip_runtime.h>
#include <math.h>

// ---------------------------------------------------------------------------
// EpisodicMemory pipeline for MI455X (gfx1250, wave32).
// Memory-bound GEMV workload (~290MB traffic, ~12.5us HBM floor @ 23.3TB/s).
// WMMA f32 16x16x4 used for the 16-row-tiled stages (sims + decoder GEMM).
// enc staging uses the gfx1250 async global->LDS path (ASYNCcnt).
// ---------------------------------------------------------------------------

typedef __attribute__((ext_vector_type(2))) float v2f;
typedef __attribute__((ext_vector_type(8))) float v8f;

#define DIM   256
#define WIN   100
#define NEP   10000
#define KIN   (DIM * WIN)     // 25600
#define H1N   (DIM * 4)       // 1024
#define H2N   (DIM * 2)       // 512
#define ENCN  (16 * DIM)      // 4096 (COMP*DIM)
#define KTOP  3
#define KCH   512             // split-K chunk for GEMV1
#define NKCH  (KIN / KCH)     // 50

__device__ __forceinline__ float gelu_exact(float x) {
    // torch nn.GELU default: 0.5*x*(1+erf(x/sqrt(2)))
    return 0.5f * x * (1.0f + erff(x * 0.70710678118654752440f));
}

__device__ __forceinline__ v8f wmma_f32_16x16x4(v2f a, v2f b, v8f c) {
    // 8-arg form: (neg_a, A, neg_b, B, c_mod, C, reuse_a, reuse_b)
    return __builtin_amdgcn_wmma_f32_16x16x4_f32(false, a, false, b,
                                                 (short)0, c, false, false);
}

// ---------------------------------------------------------------------------
// Kernel A: split-K GEMV  q[25600] @ W1[25600,1024] -> part1[50][1024]
// Deterministic (each partial written exactly once; reduced in fixed order).
// ---------------------------------------------------------------------------
__global__ __launch_bounds__(256) void k_gemv1(const float* __restrict__ q,
                                               const float* __restrict__ W1,
                                               float* __restrict__ part1) {
    const int j  = blockIdx.x * 256 + threadIdx.x;   // output column
    const int k0 = blockIdx.y * KCH;
    float acc = 0.0f;
    for (int k = k0; k < k0 + KCH; ++k)
        acc += q[k] * W1[(size_t)k * H1N + j];       // coalesced over j
    part1[(size_t)blockIdx.y * H1N + j] = acc;
}

// ---------------------------------------------------------------------------
// Kernel B: h1 = LN(gelu(sum(part1) + b1))  over 1024 elements, one WG.
// ---------------------------------------------------------------------------
__global__ __launch_bounds__(1024) void k_h1_ln(const float* __restrict__ part1,
                                                const float* __restrict__ b1,
                                                const float* __restrict__ g1,
                                                const float* __restrict__ be1,
                                                float* __restrict__ h1) {
    __shared__ float red[H1N];
    const int j = threadIdx.x;
    float acc = b1[j];
    for (int p = 0; p < NKCH; ++p) acc += part1[(size_t)p * H1N + j];
    const float x = gelu_exact(acc);

    red[j] = x; __syncthreads();
    for (int s = H1N / 2; s > 0; s >>= 1) {
        if (j < s) red[j] += red[j + s];
        __syncthreads();
    }
    const float mu = red[0] * (1.0f / (float)H1N);
    __syncthreads();
    const float d = x - mu;
    red[j] = d * d; __syncthreads();
    for (int s = H1N / 2; s > 0; s >>= 1) {
        if (j < s) red[j] += red[j + s];
        __syncthreads();
    }
    const float var = red[0] * (1.0f / (float)H1N);
    h1[j] = d * rsqrtf(var + 1e-5f) * g1[j] + be1[j];
}

// ---------------------------------------------------------------------------
// Kernel C: h2raw = h1[1024] @ W2[1024,512]   (bias applied in LN kernel)
// ---------------------------------------------------------------------------
__global__ __launch_bounds__(256) void k_gemv2(const float* __restrict__ h1,
                                               const float* __restrict__ W2,
                                               float* __restrict__ h2raw) {
    const int j = blockIdx.x * 256 + threadIdx.x;
    float acc = 0.0f;
    for (int k = 0; k < H1N; ++k)
        acc += h1[k] * W2[(size_t)k * H2N + j];
    h2raw[j] = acc;
}

// ---------------------------------------------------------------------------
// Generic per-row LN(gelu(raw + bias)) for rows of width 512.
// blockIdx.x = row, blockDim.x = 512.
// ---------------------------------------------------------------------------
__global__ __launch_bounds__(512) void k_gelu_ln512(const float* __restrict__ raw,
                                                    const float* __restrict__ bias,
                                                    const float* __restrict__ g,
                                                    const float* __restrict__ b,
                                                    float* __restrict__ out) {
    __shared__ float red[H2N];
    const int row = blockIdx.x;
    const int j = threadIdx.x;
    const float x = gelu_exact(raw[(size_t)row * H2N + j] + bias[j]);

    red[j] = x; __syncthreads();
    for (int s = H2N / 2; s > 0; s >>= 1) {
        if (j < s) red[j] += red[j + s];
        __syncthreads();
    }
    const float mu = red[0] * (1.0f / (float)H2N);
    __syncthreads();
    const float d = x - mu;
    red[j] = d * d; __syncthreads();
    for (int s = H2N / 2; s > 0; s >>= 1) {
        if (j < s) red[j] += red[j + s];
        __syncthreads();
    }
    const float var = red[0] * (1.0f / (float)H2N);
    out[(size_t)row * H2N + j] = d * rsqrtf(var + 1e-5f) * g[j] + b[j];
}

// ---------------------------------------------------------------------------
// Kernel E: enc = h2[512] @ W3[512,4096] + b3
// ---------------------------------------------------------------------------
__global__ __launch_bounds__(256) void k_gemv3(const float* __restrict__ h2,
                                               const float* __restrict__ W3,
                                               const float* __restrict__ b3,
                                               float* __restrict__ enc) {
    const int j = blockIdx.x * 256 + threadIdx.x;
    float acc = b3[j];
    for (int k = 0; k < H2N; ++k)
        acc += h2[k] * W3[(size_t)k * ENCN + j];
    enc[j] = acc;
}

// ---------------------------------------------------------------------------
// Kernel F: qn = max(||enc||, 1e-8)
// ---------------------------------------------------------------------------
__global__ __launch_bounds__(256) void k_qnorm(const float* __restrict__ enc,
                                               float* __restrict__ qn) {
    __shared__ float red[256];
    float s = 0.0f;
    for (int i = threadIdx.x; i < ENCN; i += 256) { float v = enc[i]; s += v * v; }
    red[threadIdx.x] = s; __syncthreads();
    for (int st = 128; st > 0; st >>= 1) {
        if (threadIdx.x < st) red[threadIdx.x] += red[threadIdx.x + st];
        __syncthreads();
    }
    if (threadIdx.x == 0) qn[0] = fmaxf(sqrtf(red[0]), 1e-8f);
}

// ---------------------------------------------------------------------------
// Kernel G: sims[e] = (episodes[e] . enc) / (qn * max(||episodes[e]||,1e-8))
// One wave per 16-episode tile, V_WMMA_F32_16X16X4_F32 accumulation over
// K=4096.  enc is staged in LDS with the gfx1250 async global->LDS engine
// (GLOBAL_LOAD_ASYNC_TO_LDS_B128, ASYNCcnt) and read via ds_load_b64
// broadcast.  Row norms are fused into the same stream: lane L and L+16
// cover complementary K positions of row L (A layout: K pairs {0,1} low
// half-wave, {2,3} high), so shfl_xor(16) gives the full sum of squares —
// no second 164MB pass.
// ---------------------------------------------------------------------------
__global__ __launch_bounds__(256) void k_sims_wmma(const float* __restrict__ ep,
                                                   const float* __restrict__ enc,
                                                   const float* __restrict__ qn_p,
                                                   float* __restrict__ sims) {
    __shared__ __align__(16) float s_enc[ENCN];

    // Async-stage enc (16KB) into LDS: each thread issues one b128 async copy
    // per 1024-float round (4 rounds).  LDS flat addresses map addr[31:0]
    // directly to the wave-relative LDS offset (ISA 10.2), so truncating the
    // generic pointer yields the LDS-offset VGPR the instruction expects.
    {
        const int i0 = threadIdx.x * 4;          // 16B per thread per round
#pragma unroll
        for (int r = 0; r < ENCN; r += 1024) {
            const int idx = r + i0;
            const unsigned lds_off = (unsigned)(size_t)(const void*)&s_enc[idx];
            const float* gp = enc + idx;
            asm volatile("global_load_async_to_lds_b128 %0, %1, off"
                         :: "v"(lds_off), "v"(gp)
                         : "memory");
        }
        asm volatile("s_wait_asynccnt 0x0" ::: "memory");
    }
    __syncthreads();

    const int wave = threadIdx.x >> 5;
    const int lane = threadIdx.x & 31;
    const int tile = blockIdx.x * 8 + wave;          // 625 tiles of 16 episodes
    if (tile >= NEP / 16) return;                    // wave-uniform: EXEC stays full

    const int m    = lane & 15;                      // A-matrix row for this lane
    const int koff = (lane >> 4) << 1;               // K sub-offset: 0 or 2
    const float* __restrict__ arow =
        ep + (size_t)(tile * 16 + m) * ENCN + koff;

    v8f acc = {};                                    // C = 0
    float sq = 0.0f;                                 // fused row-norm partial
    for (int k = 0; k < ENCN; k += 4) {
        v2f a = *(const v2f*)(arow + k);             // A: rows striped over lanes
        v2f b = *(const v2f*)(s_enc + k + koff);     // B: enc replicated over N
        sq += a.x * a.x + a.y * a.y;
        if ((k & 63) == 0)
            __builtin_prefetch(arow + k + 256, 0, 3); // global_prefetch_b8
        acc = wmma_f32_16x16x4(a, b, acc);
    }

    // combine complementary half-rows -> full ||row||^2 on every lane
    const float rn2 = sq + __shfl_xor(sq, 16, 32);
    const float qn  = qn_p[0];

    // D layout: VGPR v holds M=v (lanes 0-15) / M=v+8 (lanes 16-31); all N equal.
    // Lanes with N==0 (lane 0 and lane 16) write the 16 similarities.
#pragma unroll
    for (int v = 0; v < 8; ++v) {
        const int rowsel = ((lane >> 4) << 3) + v;           // 0..7 or 8..15
        const float n2 = __shfl(rn2, rowsel, 32);
        const float en = fmaxf(sqrtf(n2), 1e-8f);
        const float sv = acc[v] / (qn * en);
        if ((lane & 15) == 0)
            sims[tile * 16 + rowsel] = sv;
    }
}

// ---------------------------------------------------------------------------
// Kernel H: 3-pass argmax top-3 (lowest index wins ties, matching lax.top_k)
// ---------------------------------------------------------------------------
__global__ __launch_bounds__(256) void k_top3(const float* __restrict__ sims,
                                              int* __restrict__ topk) {
    __shared__ float sv[256];
    __shared__ int   si[256];
    __shared__ int   ssel[KTOP];
    for (int p = 0; p < KTOP; ++p) {
        float bv = -INFINITY; int bi = 0x7fffffff;
        for (int i = threadIdx.x; i < NEP; i += 256) {
            bool skip = false;
            for (int t = 0; t < p; ++t) if (ssel[t] == i) skip = true;
            const float v = sims[i];
            if (!skip && (v > bv || (v == bv && i < bi))) { bv = v; bi = i; }
        }
        sv[threadIdx.x] = bv; si[threadIdx.x] = bi;
        __syncthreads();
        for (int s = 128; s > 0; s >>= 1) {
            if (threadIdx.x < s) {
                const float ov = sv[threadIdx.x + s];
                const int   oi = si[threadIdx.x + s];
                if (ov > sv[threadIdx.x] ||
                    (ov == sv[threadIdx.x] && oi < si[threadIdx.x])) {
                    sv[threadIdx.x] = ov; si[threadIdx.x] = oi;
                }
            }
            __syncthreads();
        }
        if (threadIdx.x == 0) { ssel[p] = si[0]; topk[p] = si[0]; }
        __syncthreads();
    }
}

// ---------------------------------------------------------------------------
// Kernel I: dpre[3,512] = episodes[topk] @ Wd1[4096,512]  (WMMA, M padded 16).
// Rows 3..15 are zeroed by a multiply (uniform control flow keeps EXEC full
// for every WMMA issue).  32 waves, one 16-column tile each.
// ---------------------------------------------------------------------------
__global__ __launch_bounds__(256) void k_dec1_wmma(const float* __restrict__ ep,
                                                   const int* __restrict__ topk,
                                                   const float* __restrict__ Wd1,
                                                   float* __restrict__ dpre) {
    const int wave  = threadIdx.x >> 5;
    const int lane  = threadIdx.x & 31;
    const int tile  = blockIdx.x * 8 + wave;   // 32 tiles * 16 cols = 512
    const int nbase = tile * 16;
    const int m     = lane & 15;
    const int koff  = (lane >> 4) << 1;

    const int   mrow  = (m < KTOP) ? m : (KTOP - 1);
    const float valid = (m < KTOP) ? 1.0f : 0.0f;
    const float* __restrict__ arow =
        ep + (size_t)topk[mrow] * ENCN + koff;
    const int nc = nbase + m;                  // B column supplied by this lane

    v8f acc = {};
    for (int k = 0; k < ENCN; k += 4) {
        v2f a = *(const v2f*)(arow + k);
        a *= valid;                            // pad rows 3..15 with zeros
        v2f b;
        b.x = Wd1[(size_t)(k + koff)     * H2N + nc];
        b.y = Wd1[(size_t)(k + 1 + koff) * H2N + nc];
        acc = wmma_f32_16x16x4(a, b, acc);
    }
    // D: VGPR v = row v for lanes 0-15; only rows 0..2 are real.
    if (lane < 16) {
#pragma unroll
        for (int v = 0; v < KTOP; ++v)
            dpre[(size_t)v * H2N + nbase + lane] = acc[v];
    }
}

// ---------------------------------------------------------------------------
// Kernel K: out[256] = mean_r(dact[r]) @ Wd2[512,256] + bd2
// (mean commutes with the linear layer; fold it first.)
// ---------------------------------------------------------------------------
__global__ __launch_bounds__(256) void k_out(const float* __restrict__ dact,
                                             const float* __restrict__ Wd2,
                                             const float* __restrict__ bd2,
                                             float* __restrict__ out) {
    __shared__ float sv[H2N];
    const int j = threadIdx.x;
    for (int k = j; k < H2N; k += 256)
        sv[k] = (dact[k] + dact[H2N + k] + dact[2 * H2N + k]) * (1.0f / 3.0f);
    __syncthreads();
    float acc = bd2[j];
    for (int k = 0; k < H2N; ++k)
        acc += sv[k] * Wd2[(size_t)k * DIM + j];
    out[j] = acc;
}

// ---------------------------------------------------------------------------
// Host-side launch
// ---------------------------------------------------------------------------
extern "C" void kernel_launch(void* const* d_in, const int* in_sizes, int n_in,
                              void* d_out, int out_size, void* d_ws, size_t ws_size,
                              hipStream_t stream) {
    (void)in_sizes; (void)n_in; (void)out_size; (void)ws_size;

    const float* buf = (const float*)d_in[0];   // buffer_states [100,256]
    const float* ep  = (const float*)d_in[1];   // episodes_encoded [10000,4096]
    const float* W1  = (const float*)d_in[2];
    const float* b1  = (const float*)d_in[3];
    const float* g1  = (const float*)d_in[4];
    const float* be1 = (const float*)d_in[5];
    const float* W2  = (const float*)d_in[6];
    const float* b2  = (const float*)d_in[7];
    const float* g2  = (const float*)d_in[8];
    const float* be2 = (const float*)d_in[9];
    const float* W3  = (const float*)d_in[10];
    const float* b3  = (const float*)d_in[11];
    const float* Wd1 = (const float*)d_in[12];
    const float* bd1 = (const float*)d_in[13];
    const float* gd  = (const float*)d_in[14];
    const float* bed = (const float*)d_in[15];
    const float* Wd2 = (const float*)d_in[16];
    const float* bd2 = (const float*)d_in[17];
    // d_in[18] = k, fixed at 3 per setup_inputs
    float* out = (float*)d_out;

    float* ws    = (float*)d_ws;
    float* part1 = ws;                       // 50*1024
    float* h1    = part1 + NKCH * H1N;       // 1024
    float* h2raw = h1 + H1N;                 // 512
    float* h2    = h2raw + H2N;              // 512
    float* enc   = h2 + H2N;                 // 4096
    float* qn    = enc + ENCN;               // 1 (padded to 4)
    float* sims  = qn + 4;                   // 10000
    int*   topk  = (int*)(sims + NEP);       // 3 (padded to 4)
    float* dpre  = sims + NEP + 4;           // 3*512
    float* dact  = dpre + KTOP * H2N;        // 3*512

    // Encoder
    k_gemv1     <<<dim3(H1N / 256, NKCH), 256, 0, stream>>>(buf, W1, part1);
    k_h1_ln     <<<1, H1N, 0, stream>>>(part1, b1, g1, be1, h1);
    k_gemv2     <<<H2N / 256, 256, 0, stream>>>(h1, W2, h2raw);
    k_gelu_ln512<<<1, H2N, 0, stream>>>(h2raw, b2, g2, be2, h2);
    k_gemv3     <<<ENCN / 256, 256, 0, stream>>>(h2, W3, b3, enc);
    k_qnorm     <<<1, 256, 0, stream>>>(enc, qn);
    // Similarity + fused episode norms (WMMA + async global->LDS staging)
    k_sims_wmma <<<(NEP / 16 + 7) / 8, 256, 0, stream>>>(ep, enc, qn, sims);
    k_top3      <<<1, 256, 0, stream>>>(sims, topk);
    // Decoder (WMMA GEMM, M padded to 16)
    k_dec1_wmma <<<(H2N / 16) / 8, 256, 0, stream>>>(ep, topk, Wd1, dpre);
    k_gelu_ln512<<<KTOP, H2N, 0, stream>>>(dpre, bd1, gd, bed, dact);
    k_out       <<<1, 256, 0, stream>>>(dact, Wd2, bd2, out);
}